// StarCrossFusionModel_86955907875127
// MI455X (gfx1250) — compile-verified
//
#include <hip/hip_runtime.h>
#include <hip/hip_bf16.h>
#include <math.h>

typedef __attribute__((ext_vector_type(2))) float v2f;
typedef __attribute__((ext_vector_type(8))) float v8f;

#define B_SZ 4096
#define F_SZ 1024
#define E_SZ 16
#define D_SZ 8
#define T_SZ 1040
#define L_SZ 3
#define MP_SZ 4608   // domain-path rows padded: 8 segments each aligned up to 64

// ---------------- workspace layout (units: floats) ----------------
static constexpr size_t XC_OFF   = 0;
static constexpr size_t XC_SZ    = (size_t)B_SZ * T_SZ;          // xc (B x T)
static constexpr size_t BUFA_OFF = XC_OFF + XC_SZ;
static constexpr size_t BUFA_SZ  = (size_t)MP_SZ * 512;          // h1 (center then domain)
static constexpr size_t BUFB_OFF = BUFA_OFF + BUFA_SZ;
static constexpr size_t BUFB_SZ  = (size_t)MP_SZ * 256;          // h2
static constexpr size_t HC_OFF   = BUFB_OFF + BUFB_SZ;
static constexpr size_t HC_SZ    = (size_t)B_SZ * 128;           // center output
static constexpr size_t H3D_OFF  = HC_OFF + HC_SZ;
static constexpr size_t H3D_SZ   = (size_t)MP_SZ * 128;          // domain output (permuted)
static constexpr size_t INT_OFF  = H3D_OFF + H3D_SZ;
// int region: counts[8] | seg[8] | pad[8] | cursor[8] | perm[MP_SZ] | row_dom[MP_SZ]

// ================= LayerNorm + domain affine + concat + cross net =================
__global__ __launch_bounds__(256)
void ln_cross_kernel(const float* __restrict__ x, const int* __restrict__ dom_ids,
                     const float* __restrict__ pnw, const float* __restrict__ pnb,
                     const float* __restrict__ demb,
                     const float* __restrict__ cw, const float* __restrict__ cb,
                     float* __restrict__ xc)
{
    __shared__ float xs[T_SZ];
    __shared__ float xcs[T_SZ];
    __shared__ float red[256];

    const int b = blockIdx.x;
    const int tid = threadIdx.x;
    const int dom = dom_ids[b];
    const float* xr = x + (size_t)b * F_SZ;

    float v[4];
    float s = 0.f;
#pragma unroll
    for (int i = 0; i < 4; ++i) { v[i] = xr[tid + i * 256]; s += v[i]; }
    red[tid] = s; __syncthreads();
    for (int o = 128; o > 0; o >>= 1) { if (tid < o) red[tid] += red[tid + o]; __syncthreads(); }
    const float mean = red[0] * (1.f / F_SZ);
    __syncthreads();

    float s2 = 0.f;
#pragma unroll
    for (int i = 0; i < 4; ++i) { const float d = v[i] - mean; s2 += d * d; }
    red[tid] = s2; __syncthreads();
    for (int o = 128; o > 0; o >>= 1) { if (tid < o) red[tid] += red[tid + o]; __syncthreads(); }
    const float rstd = rsqrtf(red[0] * (1.f / F_SZ) + 1e-5f);
    __syncthreads();

#pragma unroll
    for (int i = 0; i < 4; ++i) {
        const int f = tid + i * 256;
        const float nv = (v[i] - mean) * rstd * pnw[(size_t)dom * F_SZ + f]
                       + pnb[(size_t)dom * F_SZ + f];
        xs[f] = nv; xcs[f] = nv;
    }
    if (tid < E_SZ) {
        const float e = demb[dom * E_SZ + tid];
        xs[F_SZ + tid] = e; xcs[F_SZ + tid] = e;
    }
    __syncthreads();

    for (int l = 0; l < L_SZ; ++l) {
        const float* w  = cw + l * T_SZ;
        const float* bv = cb + l * T_SZ;
        float p = 0.f;
        for (int t = tid; t < T_SZ; t += 256) p += xcs[t] * w[t];
        red[tid] = p; __syncthreads();
        for (int o = 128; o > 0; o >>= 1) { if (tid < o) red[tid] += red[tid + o]; __syncthreads(); }
        const float proj = red[0];
        __syncthreads();
        for (int t = tid; t < T_SZ; t += 256) xcs[t] = xs[t] * proj + bv[t] + xcs[t];
        __syncthreads();
    }
    for (int t = tid; t < T_SZ; t += 256) xc[(size_t)b * T_SZ + t] = xcs[t];
}

// ================= domain bucketing helpers =================
__global__ void init_kernel(int* ints)
{
    const int i = blockIdx.x * blockDim.x + threadIdx.x;
    int* perm = ints + 32;
    int* rdom = ints + 32 + MP_SZ;
    if (i < MP_SZ) { perm[i] = -1; rdom[i] = 0; }
    if (i < 8) ints[i] = 0;
}

__global__ void count_kernel(const int* __restrict__ dom_ids, int* ints)
{
    const int i = blockIdx.x * blockDim.x + threadIdx.x;
    if (i < B_SZ) atomicAdd(&ints[dom_ids[i]], 1);
}

__global__ void scan_kernel(int* ints)
{
    if (threadIdx.x == 0 && blockIdx.x == 0) {
        int* counts = ints;
        int* seg = ints + 8;
        int* pad = ints + 16;
        int* cur = ints + 24;
        int s = 0;
        for (int d = 0; d < 8; ++d) {
            seg[d] = s; cur[d] = s;
            pad[d] = (counts[d] + 63) & ~63;   // 64-row aligned segments
            s += pad[d];
        }
    }
}

__global__ void build_kernel(const int* __restrict__ dom_ids, int* ints)
{
    const int i = blockIdx.x * blockDim.x + threadIdx.x;
    if (i < B_SZ) {
        const int d = dom_ids[i];
        const int pos = atomicAdd(&ints[24 + d], 1);
        ints[32 + pos] = i;     // perm[pos] = source row
    }
}

__global__ void rowdom_kernel(int* ints)
{
    const int i = blockIdx.x * blockDim.x + threadIdx.x;
    if (i < MP_SZ) {
        const int* seg = ints + 8;
        const int* pad = ints + 16;
        int d = 0;
        for (int k = 0; k < 8; ++k)
            if (i >= seg[k] && i < seg[k] + pad[k]) d = k;
        ints[32 + MP_SZ + i] = d;
    }
}

// ================= fp32 WMMA GEMM: C[M,N] = act(A[M,K] @ W[K,N] + bias) =================
// block tile 64x128, 8 waves (2x4), wave tile 32x32 (2x2 of 16x16), K staged 16-wide in LDS
template<bool GATHER, bool RELU>
__global__ __launch_bounds__(256)
void gemm_wmma(const float* __restrict__ A, int lda,
               const float* __restrict__ W, const float* __restrict__ bias,
               float* __restrict__ C, int N, int K,
               const int* __restrict__ perm, const int* __restrict__ row_dom,
               size_t wstride, int bstride)
{
    __shared__ float Asl[64 * 18];    // [m][k], row stride 18 (even, conflict-free)
    __shared__ float Bsl[128 * 18];   // transposed [n][k]

    const int tid  = threadIdx.x;
    const int lane = tid & 31;
    const int wid  = tid >> 5;
    const int wm   = wid >> 2;        // 0..1
    const int wn   = wid & 3;         // 0..3
    const int lm   = lane & 15;
    const int lh   = lane >> 4;       // 0/1

    const int mblk = blockIdx.y * 64;
    const int nblk = blockIdx.x * 128;

    const int dom = row_dom ? row_dom[mblk] : 0;   // block-uniform (segments 64-aligned)
    const float* Wd = W + (size_t)dom * wstride;
    const float* bp = bias + (size_t)dom * bstride;

    // cooperative A-load indices: 64 rows x 16 cols, float4 per thread
    const int ar  = tid >> 2;
    const int ac4 = (tid & 3) << 2;
    int asrc;
    if (GATHER) asrc = perm[mblk + ar];
    else        asrc = mblk + ar;

    // cooperative B-load indices: 16 k x 128 n, 8 n per thread, stored transposed
    const int bk  = tid & 15;
    const int bn0 = (tid >> 4) << 3;

    v8f acc[2][2] = {};

    const int nk = K >> 4;
    for (int kc = 0; kc < nk; ++kc) {
        const int k0 = kc << 4;

        if (!GATHER || asrc >= 0) {
            const float4 av = *(const float4*)(A + (size_t)asrc * lda + k0 + ac4);
            Asl[ar * 18 + ac4 + 0] = av.x;
            Asl[ar * 18 + ac4 + 1] = av.y;
            Asl[ar * 18 + ac4 + 2] = av.z;
            Asl[ar * 18 + ac4 + 3] = av.w;
        } else {
            Asl[ar * 18 + ac4 + 0] = 0.f;
            Asl[ar * 18 + ac4 + 1] = 0.f;
            Asl[ar * 18 + ac4 + 2] = 0.f;
            Asl[ar * 18 + ac4 + 3] = 0.f;
        }

        const float* wrow = Wd + (size_t)(k0 + bk) * N + nblk + bn0;
        const float4 w0 = *(const float4*)(wrow);
        const float4 w1 = *(const float4*)(wrow + 4);
        Bsl[(bn0 + 0) * 18 + bk] = w0.x;
        Bsl[(bn0 + 1) * 18 + bk] = w0.y;
        Bsl[(bn0 + 2) * 18 + bk] = w0.z;
        Bsl[(bn0 + 3) * 18 + bk] = w0.w;
        Bsl[(bn0 + 4) * 18 + bk] = w1.x;
        Bsl[(bn0 + 5) * 18 + bk] = w1.y;
        Bsl[(bn0 + 6) * 18 + bk] = w1.z;
        Bsl[(bn0 + 7) * 18 + bk] = w1.w;
        __syncthreads();

#pragma unroll
        for (int ks = 0; ks < 16; ks += 4) {
            const v2f a0 = *(const v2f*)&Asl[(wm * 32      + lm) * 18 + ks + 2 * lh];
            const v2f a1 = *(const v2f*)&Asl[(wm * 32 + 16 + lm) * 18 + ks + 2 * lh];
            const v2f b0 = *(const v2f*)&Bsl[(wn * 32      + lm) * 18 + ks + 2 * lh];
            const v2f b1 = *(const v2f*)&Bsl[(wn * 32 + 16 + lm) * 18 + ks + 2 * lh];
            acc[0][0] = __builtin_amdgcn_wmma_f32_16x16x4_f32(false, a0, false, b0, (short)0, acc[0][0], false, false);
            acc[0][1] = __builtin_amdgcn_wmma_f32_16x16x4_f32(false, a0, false, b1, (short)0, acc[0][1], false, false);
            acc[1][0] = __builtin_amdgcn_wmma_f32_16x16x4_f32(false, a1, false, b0, (short)0, acc[1][0], false, false);
            acc[1][1] = __builtin_amdgcn_wmma_f32_16x16x4_f32(false, a1, false, b1, (short)0, acc[1][1], false, false);
        }
        __syncthreads();
    }

#pragma unroll
    for (int i = 0; i < 2; ++i) {
        const int m0 = mblk + wm * 32 + i * 16 + 8 * lh;
#pragma unroll
        for (int j = 0; j < 2; ++j) {
            const int n = nblk + wn * 32 + j * 16 + lm;
            const float bv = bp[n];
#pragma unroll
            for (int g = 0; g < 8; ++g) {
                float v = acc[i][j][g] + bv;
                if (RELU) v = fmaxf(v, 0.f);
                C[(size_t)(m0 + g) * N + n] = v;
            }
        }
    }
}

// ================= STAR fusion + final MLP + aux net + sigmoid =================
__global__ __launch_bounds__(128)
void final_kernel(const int* __restrict__ perm, const int* __restrict__ dom_ids,
                  const float* __restrict__ hc, const float* __restrict__ h3d,
                  const float* __restrict__ demb,
                  const float* __restrict__ fW1, const float* __restrict__ fb1,
                  const float* __restrict__ fW2, const float* __restrict__ fb2,
                  const float* __restrict__ aW1, const float* __restrict__ ab1,
                  const float* __restrict__ aW2, const float* __restrict__ ab2,
                  float* __restrict__ out)
{
    const int pos = blockIdx.x;
    const int b = perm[pos];
    if (b < 0) return;

    __shared__ float fs[128];
    __shared__ float h64[64];
    __shared__ float ar[32];

    const int tid = threadIdx.x;
    fs[tid] = hc[(size_t)b * 128 + tid] * tanhf(h3d[(size_t)pos * 128 + tid]);
    __syncthreads();

    if (tid < 64) {
        float s = fb1[tid];
        for (int i = 0; i < 128; ++i) s += fs[i] * fW1[i * 64 + tid];
        h64[tid] = fmaxf(s, 0.f);
    } else if (tid < 96) {
        const int j = tid - 64;
        const int dom = dom_ids[b];
        float s = ab1[j];
#pragma unroll
        for (int e = 0; e < 16; ++e) s += demb[dom * 16 + e] * aW1[e * 32 + j];
        ar[j] = fmaxf(s, 0.f) * aW2[j];
    }
    __syncthreads();

    if (tid == 0) {
        float m = fb2[0] + ab2[0];
        for (int j = 0; j < 64; ++j) m += h64[j] * fW2[j];
        for (int j = 0; j < 32; ++j) m += ar[j];
        out[b] = 1.f / (1.f + expf(-m));
    }
}

// ================= host launcher =================
extern "C" void kernel_launch(void* const* d_in, const int* in_sizes, int n_in,
                              void* d_out, int out_size, void* d_ws, size_t ws_size,
                              hipStream_t stream)
{
    const float* x    = (const float*)d_in[0];
    const int*   dids = (const int*)  d_in[1];
    const float* pnw  = (const float*)d_in[2];
    const float* pnb  = (const float*)d_in[3];
    const float* demb = (const float*)d_in[4];
    const float* cw   = (const float*)d_in[5];
    const float* cbv  = (const float*)d_in[6];
    const float* cW1  = (const float*)d_in[7];
    const float* cb1  = (const float*)d_in[8];
    const float* cW2  = (const float*)d_in[9];
    const float* cb2  = (const float*)d_in[10];
    const float* cW3  = (const float*)d_in[11];
    const float* cb3  = (const float*)d_in[12];
    const float* dW1  = (const float*)d_in[13];
    const float* db1  = (const float*)d_in[14];
    const float* dW2  = (const float*)d_in[15];
    const float* db2  = (const float*)d_in[16];
    const float* dW3  = (const float*)d_in[17];
    const float* db3  = (const float*)d_in[18];
    const float* fW1  = (const float*)d_in[19];
    const float* fb1  = (const float*)d_in[20];
    const float* fW2  = (const float*)d_in[21];
    const float* fb2  = (const float*)d_in[22];
    const float* aW1  = (const float*)d_in[23];
    const float* ab1  = (const float*)d_in[24];
    const float* aW2  = (const float*)d_in[25];
    const float* ab2  = (const float*)d_in[26];
    float* out = (float*)d_out;

    float* wsf  = (float*)d_ws;
    float* xc   = wsf + XC_OFF;
    float* bufA = wsf + BUFA_OFF;
    float* bufB = wsf + BUFB_OFF;
    float* hc   = wsf + HC_OFF;
    float* h3d  = wsf + H3D_OFF;
    int*   ints = (int*)(wsf + INT_OFF);
    int*   perm = ints + 32;
    int*   rdom = ints + 32 + MP_SZ;

    (void)in_sizes; (void)n_in; (void)out_size; (void)ws_size;

    // bucketing + LN/cross
    init_kernel<<<(MP_SZ + 255) / 256, 256, 0, stream>>>(ints);
    ln_cross_kernel<<<B_SZ, 256, 0, stream>>>(x, dids, pnw, pnb, demb, cw, cbv, xc);
    count_kernel<<<(B_SZ + 255) / 256, 256, 0, stream>>>(dids, ints);
    scan_kernel<<<1, 1, 0, stream>>>(ints);
    build_kernel<<<(B_SZ + 255) / 256, 256, 0, stream>>>(dids, ints);
    rowdom_kernel<<<(MP_SZ + 255) / 256, 256, 0, stream>>>(ints);

    // center MLP
    gemm_wmma<false, true ><<<dim3(4, 64), 256, 0, stream>>>(xc,   T_SZ, cW1, cb1, bufA, 512, T_SZ, nullptr, nullptr, 0, 0);
    gemm_wmma<false, true ><<<dim3(2, 64), 256, 0, stream>>>(bufA, 512,  cW2, cb2, bufB, 256, 512,  nullptr, nullptr, 0, 0);
    gemm_wmma<false, false><<<dim3(1, 64), 256, 0, stream>>>(bufB, 256,  cW3, cb3, hc,   128, 256,  nullptr, nullptr, 0, 0);

    // domain nets: grouped GEMM over permuted, domain-sorted rows (only the selected net per row)
    gemm_wmma<true,  true ><<<dim3(4, MP_SZ / 64), 256, 0, stream>>>(xc,   T_SZ, dW1, db1, bufA, 512, T_SZ, perm, rdom, (size_t)T_SZ * 512, 512);
    gemm_wmma<false, true ><<<dim3(2, MP_SZ / 64), 256, 0, stream>>>(bufA, 512,  dW2, db2, bufB, 256, 512,  nullptr, rdom, (size_t)512 * 256, 256);
    gemm_wmma<false, false><<<dim3(1, MP_SZ / 64), 256, 0, stream>>>(bufB, 256,  dW3, db3, h3d, 128, 256,  nullptr, rdom, (size_t)256 * 128, 128);

    // fusion + heads
    final_kernel<<<MP_SZ, 128, 0, stream>>>(perm, dids, hc, h3d, demb,
                                            fW1, fb1, fW2, fb2, aW1, ab1, aW2, ab2, out);
}